// Loss_24283745092302
// MI455X (gfx1250) — compile-verified
//
#include <hip/hip_runtime.h>
#include <cstdint>

// ---------------- problem constants ----------------
#define S_GRID 7
#define BATCH 16384
#define D_FEAT 30                               // B*5 + C = 30
#define NCELLS (BATCH * S_GRID * S_GRID)        // 802816
#define CELLS_PER_BLOCK 256
#define NBLOCKS (NCELLS / CELLS_PER_BLOCK)      // 3136 exactly
#define TILE_ELEMS (CELLS_PER_BLOCK * D_FEAT)   // 7680 floats = 30720 B

typedef __attribute__((ext_vector_type(4))) unsigned int v4u;
typedef __attribute__((ext_vector_type(8))) int          v8i;
typedef __attribute__((ext_vector_type(4))) int          v4i;

// ---------------- TDM: 1-D tile of TILE_ELEMS f32 from global -> LDS -------
// D# layout per CDNA5 ISA.
__device__ __forceinline__ void tdm_load_tile(const float* gsrc, unsigned lds_byte_addr) {
  unsigned long long ga = (unsigned long long)(uintptr_t)gsrc;

  v4u g0;
  g0[0] = 1u;                                   // count=1 (valid), user mode
  g0[1] = lds_byte_addr;                        // lds_addr [63:32]
  g0[2] = (unsigned)ga;                         // global_addr lo32
  g0[3] = ((unsigned)(ga >> 32) & 0x01FFFFFFu)  // global_addr [56:32]
        | 0x80000000u;                          // type=2 ("image") at [127:126]

  v8i g1;
  g1[0] = 0x00020000;                           // wg_mask=0, data_size=2 (4 B)
  g1[1] = (int)(((unsigned)TILE_ELEMS & 0xFFFFu) << 16); // tensor_dim0 lo16 @[63:48]
  g1[2] = (int)((((unsigned)TILE_ELEMS >> 16) & 0xFFFFu) // tensor_dim0 hi16 @[79:64]
        | (1u << 16));                          // tensor_dim1 lo16 = 1 @[95:80]
  g1[3] = (int)((unsigned)TILE_ELEMS << 16);    // tensor_dim1 hi=0, tile_dim0 @[127:112]
  g1[4] = (int)(1u | (1u << 16));               // tile_dim1=1, tile_dim2=1
  g1[5] = TILE_ELEMS;                           // tensor_dim0_stride lo32
  g1[6] = 0;                                    // stride0 hi16 | stride1 lo16
  g1[7] = 0;                                    // stride1 hi32

  v4i g2;                                       // group2 (normal mode)
  g2[0] = 1;                                    // tensor_dim2 = 1
  g2[1] = 1;                                    // tensor_dim3 = 1
  g2[2] = 0;                                    // tensor_dim2_stride lo32
  g2[3] = (int)(1u << 16);                      // stride hi=0, tile_dim3=1

  v4i g3;                                       // group3
  g3[0] = 0;                                    // tensor_dim3_stride lo32
  g3[1] = (int)(1u << 16);                      // stride3 hi=0, tensor_dim4 lo16=1
  g3[2] = (int)(1u << 16);                      // tensor_dim4 hi=0, tile_dim4=1
  g3[3] = 0;

#if __clang_major__ >= 23
  v8i g4 = {0, 0, 0, 0, 0, 0, 0, 0};
  __builtin_amdgcn_tensor_load_to_lds(g0, g1, g2, g3, g4, 0);
#else
  __builtin_amdgcn_tensor_load_to_lds(g0, g1, g2, g3, 0);
#endif
}

__device__ __forceinline__ float wave_reduce_add(float v) {
#pragma unroll
  for (int off = 16; off > 0; off >>= 1)
    v += __shfl_down(v, off, 32);
  return v;
}

// ---------------- pass 1: per-block partial losses ----------------
__global__ void __launch_bounds__(CELLS_PER_BLOCK)
yolo_loss_pass1(const float* __restrict__ pred, const float* __restrict__ tgt,
                float* __restrict__ ws) {
  __shared__ float s_pred[TILE_ELEMS];
  __shared__ float s_tgt[TILE_ELEMS];
  __shared__ float s_red[CELLS_PER_BLOCK / 32][5];

  const int tid  = threadIdx.x;
  const int wv   = tid >> 5;
  const int lane = tid & 31;
  const size_t tile_base = (size_t)blockIdx.x * TILE_ELEMS;

  // TDM bulk stage: wave0 -> pred tile, wave1 -> target tile.
  // TDM ignores EXEC; one issue per wave. Each issuing wave drains its own
  // TENSORcnt before the workgroup barrier makes LDS visible to all waves.
  if (wv == 0) {
    tdm_load_tile(pred + tile_base, (unsigned)(uintptr_t)&s_pred[0]);
    __builtin_amdgcn_s_wait_tensorcnt(0);
  } else if (wv == 1) {
    tdm_load_tile(tgt + tile_base, (unsigned)(uintptr_t)&s_tgt[0]);
    __builtin_amdgcn_s_wait_tensorcnt(0);
  }
  __syncthreads();

  // Per-cell data: 30 f32, base tid*120 B (8-B aligned) -> conflict-free ds_load_b64.
  float pv[D_FEAT], tv[D_FEAT];
  {
    const float2* pp = (const float2*)&s_pred[tid * D_FEAT];
    const float2* tp = (const float2*)&s_tgt[tid * D_FEAT];
#pragma unroll
    for (int i = 0; i < D_FEAT / 2; ++i) {
      float2 a = pp[i]; pv[2 * i] = a.x; pv[2 * i + 1] = a.y;
      float2 b = tp[i]; tv[2 * i] = b.x; tv[2 * i + 1] = b.y;
    }
  }

  const float t4 = tv[4];
  const bool obj   = (t4 > 0.0f);
  const bool noobj = (t4 == 0.0f);

  // no-object confidence loss (conf indices 4, 9)
  float l_noobj;
  {
    float d0 = pv[4] - tv[4];
    float d1 = pv[9] - tv[9];
    l_noobj = noobj ? (d0 * d0 + d1 * d1) : 0.0f;
  }

  // class loss (indices 10..29)
  float l_class = 0.0f;
#pragma unroll
  for (int j = 10; j < D_FEAT; ++j) {
    float d = pv[j] - tv[j];
    l_class += d * d;
  }
  l_class = obj ? l_class : 0.0f;

  // target corners from target box 0
  const float txc = tv[0] / 7.0f, tyc = tv[1] / 7.0f;
  const float tc0 = txc - 0.5f * tv[2], tc1 = tyc - 0.5f * tv[3];
  const float tc2 = txc + 0.5f * tv[2], tc3 = tyc + 0.5f * tv[3];
  const float area_t = (tc2 - tc0) * (tc3 - tc1);

  float iou[2];
#pragma unroll
  for (int b = 0; b < 2; ++b) {
    const int o = 5 * b;
    float pxc = pv[o + 0] / 7.0f, pyc = pv[o + 1] / 7.0f;
    float pc0 = pxc - 0.5f * pv[o + 2], pc1 = pyc - 0.5f * pv[o + 3];
    float pc2 = pxc + 0.5f * pv[o + 2], pc3 = pyc + 0.5f * pv[o + 3];
    float lt0 = fmaxf(pc0, tc0), lt1 = fmaxf(pc1, tc1);
    float rb0 = fminf(pc2, tc2), rb1 = fminf(pc3, tc3);
    float w = fmaxf(rb0 - lt0, 0.0f), h = fmaxf(rb1 - lt1, 0.0f);
    float inter = w * h;
    float area_p = (pc2 - pc0) * (pc3 - pc1);
    iou[b] = inter / (area_p + area_t - inter);
  }
  const float max_iou = fmaxf(iou[0], iou[1]);
  const int best = (iou[1] > iou[0]) ? 1 : 0;   // jnp.argmax tie -> first index

  float l_xy = 0.0f, l_wh = 0.0f, l_obj = 0.0f;
#pragma unroll
  for (int b = 0; b < 2; ++b) {
    const bool resp = obj && (b == best);
    const int o = 5 * b;
    float dx = pv[o + 0] - tv[o + 0];
    float dy = pv[o + 1] - tv[o + 1];
    float dw = sqrtf(pv[o + 2]) - sqrtf(tv[o + 2]);
    float dh = sqrtf(pv[o + 3]) - sqrtf(tv[o + 3]);
    float db = pv[o + 4] - max_iou;
    if (resp) {
      l_xy  += dx * dx + dy * dy;
      l_wh  += dw * dw + dh * dh;
      l_obj += db * db;
    }
  }

  // deterministic block reduction: wave32 shuffle tree -> LDS -> 5 threads
  float vals[5] = {l_xy, l_wh, l_obj, l_noobj, l_class};
#pragma unroll
  for (int c = 0; c < 5; ++c) {
    float r = wave_reduce_add(vals[c]);
    if (lane == 0) s_red[wv][c] = r;
  }
  __syncthreads();
  if (tid < 5) {
    float s = 0.0f;
#pragma unroll
    for (int w = 0; w < CELLS_PER_BLOCK / 32; ++w) s += s_red[w][tid];
    ws[(size_t)blockIdx.x * 5 + tid] = s;
  }
}

// ---------------- pass 2: deterministic final fold ----------------
__global__ void __launch_bounds__(256)
yolo_loss_pass2(const float* __restrict__ ws, float* __restrict__ out, int nblocks) {
  __shared__ float red[256];
  const float inv_batch = 1.0f / (float)BATCH;
  for (int c = 0; c < 5; ++c) {
    float s = 0.0f;
    for (int i = threadIdx.x; i < nblocks; i += 256) s += ws[(size_t)i * 5 + c];
    red[threadIdx.x] = s;
    __syncthreads();
    for (int stride = 128; stride > 0; stride >>= 1) {
      if (threadIdx.x < (unsigned)stride) red[threadIdx.x] += red[threadIdx.x + stride];
      __syncthreads();
    }
    if (threadIdx.x == 0) out[c] = red[0] * inv_batch;
    __syncthreads();
  }
}

// ---------------- launch ----------------
extern "C" void kernel_launch(void* const* d_in, const int* in_sizes, int n_in,
                              void* d_out, int out_size, void* d_ws, size_t ws_size,
                              hipStream_t stream) {
  const float* pred = (const float*)d_in[0];   // pred_tensor  (16384,7,7,30) f32
  const float* tgt  = (const float*)d_in[1];   // target_tensor(16384,7,7,30) f32
  float* ws  = (float*)d_ws;                   // NBLOCKS*5 f32 partials (62,720 B)
  float* out = (float*)d_out;                  // 5 f32: xy, wh, obj, noobj, class

  yolo_loss_pass1<<<NBLOCKS, CELLS_PER_BLOCK, 0, stream>>>(pred, tgt, ws);
  yolo_loss_pass2<<<1, 256, 0, stream>>>(ws, out, NBLOCKS);
}